// GraphNeuralNetworkWithSpatialInfo_65747359367323
// MI455X (gfx1250) — compile-verified
//
#include <hip/hip_runtime.h>
#include <math.h>

// ---------------------------------------------------------------------------
// GCN block for gfx1250 (MI455X):
//  - GEMM via V_WMMA_F32_16X16X4_F32 with double-buffered
//    GLOBAL_LOAD_ASYNC_TO_LDS_B128 staging (ASYNCcnt / s_wait_asynccnt)
//  - edge aggregation via global_atomic_add_f32
//  - wave32 layernorm reductions
// ---------------------------------------------------------------------------

typedef float v2f __attribute__((ext_vector_type(2)));
typedef float v8f __attribute__((ext_vector_type(8)));

#define D 512           // feature dim (in = hid = out = 512)
#define KC 32           // K chunk staged in LDS
#define TROWS 128       // C rows per workgroup (8 waves x 16 rows)
#define TCOLS 64        // C cols per workgroup (each wave: 16x64 strip)
#define APAD 8          // row pad (keeps rows 16B-aligned, spreads banks)
#define BPAD 8

__device__ __forceinline__ float gelu_exact(float x) {
    return 0.5f * x * (1.0f + erff(x * 0.70710678118654752440f));
}

// Async global -> LDS copy of 16 bytes per lane (CDNA5, ASYNCcnt-tracked).
__device__ __forceinline__ void async_b128(unsigned int lds_off, const void* gaddr) {
    asm volatile("global_load_async_to_lds_b128 %0, %1, off"
                 :: "v"(lds_off), "v"(gaddr)
                 : "memory");
}

// ---------------------------------------------------------------------------
// Degree / normalization
// ---------------------------------------------------------------------------
__global__ void deg_init_kernel(int* __restrict__ deg, int n) {
    int i = blockIdx.x * blockDim.x + threadIdx.x;
    if (i < n) deg[i] = 1;   // self-loop
}

__global__ void deg_count_kernel(int* __restrict__ deg, const int* __restrict__ dst, int e) {
    int i = blockIdx.x * blockDim.x + threadIdx.x;
    if (i < e) atomicAdd(&deg[dst[i]], 1);
}

__global__ void dinv_kernel(const int* __restrict__ deg, float* __restrict__ dinv, int n) {
    int i = blockIdx.x * blockDim.x + threadIdx.x;
    if (i < n) dinv[i] = rsqrtf((float)deg[i]);  // deg >= 1 always
}

// ---------------------------------------------------------------------------
// C[M x 512] = A[M x 512] @ B[512 x 512], f32, WMMA 16x16x4 f32.
// Workgroup: 256 threads = 8 waves; wave w computes a 16x64 strip of C.
// K is streamed in KC-chunks, double-buffered in LDS via async-to-LDS copies:
// each thread issues exactly 6 b128 copies per chunk (4 for A, 2 for B), so
// "s_wait_asynccnt 6" after issuing the next chunk guarantees the current
// chunk has landed (async loads complete in order).
// ---------------------------------------------------------------------------
__launch_bounds__(256)
__global__ void gemm_wmma_f32_kernel(const float* __restrict__ A,
                                     const float* __restrict__ B,
                                     float* __restrict__ C, int M) {
    __shared__ __align__(16) float As[2][TROWS][KC + APAD];  // 2*128*40*4 = 40KB
    __shared__ __align__(16) float Bs[2][KC][TCOLS + BPAD];  // 2*32*72*4  = 18KB

    const int tid  = threadIdx.x;
    const int lane = tid & 31;
    const int wave = tid >> 5;                 // 0..7
    const int blkRow = blockIdx.y * TROWS;
    const int blkCol = blockIdx.x * TCOLS;

    const int l16 = lane & 15;                 // m for A-frag, n for B/C-frag
    const int hi  = lane >> 4;                 // 0 or 1: K-half / M-half selector

    v8f acc[4] = {};

    // ---- per-thread async copy descriptors (uniform, never divergent) ----
    const float* ga[4];
    unsigned int la[4];
    #pragma unroll
    for (int i = 0; i < 4; ++i) {              // A tile: 1024 float4, 4/thread
        int idx = tid + 256 * i;
        int r = idx >> 3, c4 = idx & 7;        // 8 float4 per 32-float row
        int gr = blkRow + r;
        if (gr >= M) gr = M - 1;               // clamp: no divergence; rows guarded at store
        ga[i] = A + (size_t)gr * D + c4 * 4;
        la[i] = (unsigned int)(uintptr_t)(&As[0][r][c4 * 4]);
    }
    const float* gb[2];
    unsigned int lb[2];
    #pragma unroll
    for (int i = 0; i < 2; ++i) {              // B tile: 512 float4, 2/thread
        int idx = tid + 256 * i;
        int r = idx >> 4, c4 = idx & 15;       // 16 float4 per 64-float row
        gb[i] = B + (size_t)r * D + blkCol + c4 * 4;
        lb[i] = (unsigned int)(uintptr_t)(&Bs[0][r][c4 * 4]);
    }
    const unsigned int aBufBytes = sizeof(As[0]);
    const unsigned int bBufBytes = sizeof(Bs[0]);

    auto issue_chunk = [&](int chunk, int buf) {
        const size_t aOff = (size_t)chunk * KC;       // +KC floats along K
        const size_t bOff = (size_t)chunk * KC * D;   // +KC rows of B
        #pragma unroll
        for (int i = 0; i < 4; ++i)
            async_b128(la[i] + buf * aBufBytes, ga[i] + aOff);
        #pragma unroll
        for (int i = 0; i < 2; ++i)
            async_b128(lb[i] + buf * bBufBytes, gb[i] + bOff);
    };

    const int nchunks = D / KC;                // 16
    issue_chunk(0, 0);

    for (int c = 0; c < nchunks; ++c) {
        const int cur = c & 1;
        if (c + 1 < nchunks) {
            issue_chunk(c + 1, cur ^ 1);
            asm volatile("s_wait_asynccnt 0x6" ::: "memory");  // chunk c landed
        } else {
            asm volatile("s_wait_asynccnt 0x0" ::: "memory");
        }
        __syncthreads();                       // everyone's chunk c visible

        #pragma unroll
        for (int kk = 0; kk < KC; kk += 4) {
            // A frag (16x4): lane 16*hi+m holds A[m][kk+2*hi], A[m][kk+2*hi+1]
            v2f a = *(const v2f*)(&As[cur][wave * 16 + l16][kk + 2 * hi]);
            #pragma unroll
            for (int ct = 0; ct < 4; ++ct) {
                // B frag (4x16): lane 16*hi+n holds B[kk+2*hi][n], B[kk+2*hi+1][n]
                v2f b;
                b.x = Bs[cur][kk + 2 * hi][ct * 16 + l16];
                b.y = Bs[cur][kk + 2 * hi + 1][ct * 16 + l16];
                acc[ct] = __builtin_amdgcn_wmma_f32_16x16x4_f32(
                    false, a, false, b, (short)0, acc[ct], false, false);
            }
        }
        __syncthreads();                       // buffer free for chunk c+2
    }

    // C/D layout (16x16 f32): lane 16*hi+n, VGPR v -> C[v + 8*hi][n]
    #pragma unroll
    for (int ct = 0; ct < 4; ++ct) {
        #pragma unroll
        for (int v = 0; v < 8; ++v) {
            int gr = blkRow + wave * 16 + v + 8 * hi;
            if (gr < M)
                C[(size_t)gr * D + blkCol + ct * 16 + l16] = acc[ct][v];
        }
    }
}

// ---------------------------------------------------------------------------
// out[i] = dinv[i]^2 * xw[i] + bias   (self-loop message + bias; also fully
// initializes out before the edge-atomics kernel).
// ---------------------------------------------------------------------------
__launch_bounds__(256)
__global__ void selfloop_bias_kernel(const float* __restrict__ xw,
                                     const float* __restrict__ dinv,
                                     const float* __restrict__ bias,
                                     float* __restrict__ out, int n) {
    int idx = blockIdx.x * blockDim.x + threadIdx.x;   // over n*128 float4s
    int total = n * (D / 4);
    if (idx >= total) return;
    int row = idx >> 7;
    int c   = (idx & 127) * 4;
    float di = dinv[row];
    float s  = di * di;
    float4 v = *(const float4*)(xw + (size_t)row * D + c);
    float4 b = *(const float4*)(bias + c);
    float4 o;
    o.x = s * v.x + b.x;
    o.y = s * v.y + b.y;
    o.z = s * v.z + b.z;
    o.w = s * v.w + b.w;
    *(float4*)(out + (size_t)row * D + c) = o;
}

// ---------------------------------------------------------------------------
// One wave per edge: out[dst] += dinv[src]*dinv[dst] * xw[src]
// ---------------------------------------------------------------------------
__launch_bounds__(256)
__global__ void edge_aggregate_kernel(const float* __restrict__ xw,
                                      const float* __restrict__ dinv,
                                      const int* __restrict__ src,
                                      const int* __restrict__ dst,
                                      float* __restrict__ out, int e) {
    int lane = threadIdx.x & 31;
    int edge = blockIdx.x * (blockDim.x >> 5) + (threadIdx.x >> 5);
    if (edge >= e) return;
    int s = src[edge];
    int d = dst[edge];
    float c = dinv[s] * dinv[d];
    const float* xs = xw + (size_t)s * D;
    float* od = out + (size_t)d * D;
    __builtin_prefetch(od + lane * 4, 1, 0);   // global_prefetch_b8 on dst row
    #pragma unroll
    for (int t = 0; t < 4; ++t) {
        int j = (lane + 32 * t) * 4;
        float4 v = *(const float4*)(xs + j);
        unsafeAtomicAdd(od + j + 0, c * v.x);
        unsafeAtomicAdd(od + j + 1, c * v.y);
        unsafeAtomicAdd(od + j + 2, c * v.z);
        unsafeAtomicAdd(od + j + 3, c * v.w);
    }
}

// ---------------------------------------------------------------------------
// Wave-per-row LayerNorm + (optional residual) + exact GELU, in place.
// blockDim = 256 -> 8 rows per workgroup.
// ---------------------------------------------------------------------------
__launch_bounds__(256)
__global__ void ln_act_kernel(float* __restrict__ h,
                              const float* __restrict__ gamma,
                              const float* __restrict__ beta,
                              const float* __restrict__ resid,   // may be null
                              int n, int addResid) {
    int lane = threadIdx.x & 31;
    int row  = blockIdx.x * (blockDim.x >> 5) + (threadIdx.x >> 5);
    if (row >= n) return;
    float* hr = h + (size_t)row * D;

    float4 vals[4];
    float sum = 0.f;
    #pragma unroll
    for (int t = 0; t < 4; ++t) {
        vals[t] = *(const float4*)(hr + (lane + 32 * t) * 4);
        sum += vals[t].x + vals[t].y + vals[t].z + vals[t].w;
    }
    #pragma unroll
    for (int off = 16; off > 0; off >>= 1) sum += __shfl_xor(sum, off, 32);
    float mu = sum * (1.0f / (float)D);

    float vs = 0.f;
    #pragma unroll
    for (int t = 0; t < 4; ++t) {
        float dx = vals[t].x - mu; vs += dx * dx;
        float dy = vals[t].y - mu; vs += dy * dy;
        float dz = vals[t].z - mu; vs += dz * dz;
        float dw = vals[t].w - mu; vs += dw * dw;
    }
    #pragma unroll
    for (int off = 16; off > 0; off >>= 1) vs += __shfl_xor(vs, off, 32);
    float inv = rsqrtf(vs * (1.0f / (float)D) + 1e-5f);

    const float* rr = addResid ? (resid + (size_t)row * D) : nullptr;
    #pragma unroll
    for (int t = 0; t < 4; ++t) {
        int j = (lane + 32 * t) * 4;
        float4 g = *(const float4*)(gamma + j);
        float4 b = *(const float4*)(beta + j);
        float4 o;
        o.x = (vals[t].x - mu) * inv * g.x + b.x;
        o.y = (vals[t].y - mu) * inv * g.y + b.y;
        o.z = (vals[t].z - mu) * inv * g.z + b.z;
        o.w = (vals[t].w - mu) * inv * g.w + b.w;
        if (addResid) {
            float4 r = *(const float4*)(rr + j);
            o.x += r.x; o.y += r.y; o.z += r.z; o.w += r.w;
        }
        o.x = gelu_exact(o.x);
        o.y = gelu_exact(o.y);
        o.z = gelu_exact(o.z);
        o.w = gelu_exact(o.w);
        *(float4*)(hr + j) = o;
    }
}

// ---------------------------------------------------------------------------
// Host-side orchestration
// ---------------------------------------------------------------------------
extern "C" void kernel_launch(void* const* d_in, const int* in_sizes, int n_in,
                              void* d_out, int out_size, void* d_ws, size_t ws_size,
                              hipStream_t stream) {
    // setup_inputs order:
    // 0:x 1:edge_index 2:batch 3:W1 4:b1 5:gamma1 6:beta1 7:W2 8:b2 9:gamma2 10:beta2
    const float* x      = (const float*)d_in[0];
    const int*   ei     = (const int*)d_in[1];
    const float* W1     = (const float*)d_in[3];
    const float* b1     = (const float*)d_in[4];
    const float* gamma1 = (const float*)d_in[5];
    const float* beta1  = (const float*)d_in[6];
    const float* W2     = (const float*)d_in[7];
    const float* b2     = (const float*)d_in[8];
    const float* gamma2 = (const float*)d_in[9];
    const float* beta2  = (const float*)d_in[10];

    const int N = in_sizes[0] / D;        // 30000
    const int E = in_sizes[1] / 2;        // 480000
    const int* src = ei;
    const int* dst = ei + E;

    float* out = (float*)d_out;           // N x 512

    // workspace layout
    float* xw   = (float*)d_ws;                       // N*512
    float* h    = xw + (size_t)N * D;                 // N*512
    float* dinv = h + (size_t)N * D;                  // N
    int*   deg  = (int*)(dinv + N);                   // N
    (void)ws_size; (void)n_in; (void)out_size;

    const int rowBlocks = (N + TROWS - 1) / TROWS;    // 235
    const dim3 gemmGrid(D / TCOLS, rowBlocks);        // (8, 235)
    const int slBlocks   = (N * (D / 4) + 255) / 256; // self-loop grid
    const int edgeBlocks = (E + 7) / 8;               // 1 wave per edge
    const int rowWaves   = (N + 7) / 8;               // LN: 8 rows per block

    // --- normalization coefficients ---
    deg_init_kernel<<<(N + 255) / 256, 256, 0, stream>>>(deg, N);
    deg_count_kernel<<<(E + 255) / 256, 256, 0, stream>>>(deg, dst, E);
    dinv_kernel<<<(N + 255) / 256, 256, 0, stream>>>(deg, dinv, N);

    // --- layer 1: GCNConv -> LN -> GELU ---
    gemm_wmma_f32_kernel<<<gemmGrid, 256, 0, stream>>>(x, W1, xw, N);
    selfloop_bias_kernel<<<slBlocks, 256, 0, stream>>>(xw, dinv, b1, h, N);
    edge_aggregate_kernel<<<edgeBlocks, 256, 0, stream>>>(xw, dinv, src, dst, h, E);
    ln_act_kernel<<<rowWaves, 256, 0, stream>>>(h, gamma1, beta1, nullptr, N, 0);

    // --- layer 2: GCNConv -> LN -> +residual -> GELU ---
    gemm_wmma_f32_kernel<<<gemmGrid, 256, 0, stream>>>(h, W2, xw, N);
    selfloop_bias_kernel<<<slBlocks, 256, 0, stream>>>(xw, dinv, b2, out, N);
    edge_aggregate_kernel<<<edgeBlocks, 256, 0, stream>>>(xw, dinv, src, dst, out, E);
    ln_act_kernel<<<rowWaves, 256, 0, stream>>>(out, gamma2, beta2, x, N, 1);
}